// SAGEModelDeep_28741921144896
// MI455X (gfx1250) — compile-verified
//
#include <hip/hip_runtime.h>
#include <hip/hip_bf16.h>

// ---------------------------------------------------------------------------
// GraphSAGE (3 layers, D=128) + BN + ReLU + residual + final linear (C=47)
// CDNA5 / gfx1250: fp32 WMMA (v_wmma_f32_16x16x4_f32) for all GEMMs.
// ---------------------------------------------------------------------------

#define GN 100000      // nodes
#define GE 1600000     // edges
#define GD 128         // feature dim
#define GC 47          // classes
#define BN_BLOCKS 256

typedef float v2f __attribute__((ext_vector_type(2)));
typedef float v8f __attribute__((ext_vector_type(8)));

// ---------------------------------------------------------------------------
// Zero the aggregation buffer and degree counters.
// ---------------------------------------------------------------------------
__global__ void zero_kernel(float* __restrict__ agg, float* __restrict__ deg) {
    size_t total = (size_t)GN * GD;
    for (size_t i = (size_t)blockIdx.x * blockDim.x + threadIdx.x; i < total;
         i += (size_t)gridDim.x * blockDim.x) {
        agg[i] = 0.0f;
        if (i < GN) deg[i] = 0.0f;
    }
}

// ---------------------------------------------------------------------------
// Edge scatter: one wave (32 lanes) per edge, each lane handles 4 features.
// agg[dst] += x[src]; deg[dst] += 1
// ---------------------------------------------------------------------------
__global__ __launch_bounds__(256) void scatter_kernel(
    const float* __restrict__ x, const int* __restrict__ src,
    const int* __restrict__ dst, float* __restrict__ agg,
    float* __restrict__ deg) {
    const int lane = threadIdx.x & 31;
    const int e = blockIdx.x * 8 + (threadIdx.x >> 5);
    if (e >= GE) return;  // wave-uniform
    const int s = src[e];
    const int d = dst[e];
    const float4 v = *(const float4*)(x + (size_t)s * GD + lane * 4);
    float* a = agg + (size_t)d * GD + lane * 4;
    atomicAdd(a + 0, v.x);
    atomicAdd(a + 1, v.y);
    atomicAdd(a + 2, v.z);
    atomicAdd(a + 3, v.w);
    if (lane == 0) atomicAdd(deg + d, 1.0f);
}

// ---------------------------------------------------------------------------
// deg[i] -> 1 / max(deg[i], 1)   (in place)
// ---------------------------------------------------------------------------
__global__ void rdeg_kernel(float* __restrict__ deg) {
    int i = blockIdx.x * blockDim.x + threadIdx.x;
    if (i < GN) {
        float c = deg[i];
        deg[i] = 1.0f / fmaxf(c, 1.0f);
    }
}

// ---------------------------------------------------------------------------
// SAGE GEMM:  h = x @ Wl^T + bl + (agg * rdeg) @ Wr^T
// One wave computes a 16-row x 128-col output tile via fp32 WMMA 16x16x4.
// A fragment (16x4 f32): lane<16 holds (M=lane, K=k0..k0+1), lane>=16 holds
// (M=lane-16, K=k0+2..k0+3)  -> one v2f load per lane, reused for 8 col tiles.
// B fragment (4x16 f32): lane<16 holds (N=lane, K=k0..k0+1) etc.; since we
// need B[k][n] = W[n][k] and W is row-major [out][in], each lane loads a
// K-contiguous v2f from row n of W.
// ---------------------------------------------------------------------------
__global__ __launch_bounds__(256) void sage_gemm_kernel(
    const float* __restrict__ x, const float* __restrict__ agg,
    const float* __restrict__ rdeg, const float* __restrict__ Wl,
    const float* __restrict__ bl, const float* __restrict__ Wr,
    float* __restrict__ h) {
    const int lane = threadIdx.x & 31;
    const int wave = threadIdx.x >> 5;
    const int tile = blockIdx.x * 8 + wave;  // 16-row tile index
    if (tile >= GN / 16) return;             // wave-uniform guard
    const int row0 = tile * 16;
    const bool hi = lane >= 16;
    const int l16 = lane & 15;
    const int m = row0 + l16;       // A-matrix row owned by this lane
    const int koff = hi ? 2 : 0;    // K sub-offset within the 4-wide step

    v8f zero = {};
    v8f acc[8];
#pragma unroll
    for (int t = 0; t < 8; ++t) acc[t] = zero;

    // ---- x @ Wl^T ----
    {
        const float* arow = x + (size_t)m * GD;
        for (int k0 = 0; k0 < GD; k0 += 4) {
            v2f a = *(const v2f*)(arow + k0 + koff);
#pragma unroll
            for (int t = 0; t < 8; ++t) {
                const int n = t * 16 + l16;
                v2f b = *(const v2f*)(Wl + (size_t)n * GD + k0 + koff);
                acc[t] = __builtin_amdgcn_wmma_f32_16x16x4_f32(
                    false, a, false, b, (short)0, acc[t], false, false);
            }
        }
    }
    // ---- mean @ Wr^T  (mean = agg * rdeg, formed on the fly) ----
    {
        const float rd = rdeg[m];
        const float* grow = agg + (size_t)m * GD;
        for (int k0 = 0; k0 < GD; k0 += 4) {
            v2f a = *(const v2f*)(grow + k0 + koff);
            a *= rd;
#pragma unroll
            for (int t = 0; t < 8; ++t) {
                const int n = t * 16 + l16;
                v2f b = *(const v2f*)(Wr + (size_t)n * GD + k0 + koff);
                acc[t] = __builtin_amdgcn_wmma_f32_16x16x4_f32(
                    false, a, false, b, (short)0, acc[t], false, false);
            }
        }
    }

    // ---- store: C/D layout — VGPR v holds row (v + hi*8), col = l16 ----
#pragma unroll
    for (int t = 0; t < 8; ++t) {
        const int col = t * 16 + l16;
        const float bias = bl[col];
#pragma unroll
        for (int v = 0; v < 8; ++v) {
            const int r = row0 + v + (hi ? 8 : 0);
            h[(size_t)r * GD + col] = acc[t][v] + bias;
        }
    }
}

// ---------------------------------------------------------------------------
// BatchNorm pass 1: per-block partial column sums of h and h^2 (deterministic).
// ---------------------------------------------------------------------------
__global__ __launch_bounds__(128) void bn_partial_kernel(
    const float* __restrict__ h, float* __restrict__ partial) {
    const int col = threadIdx.x;  // 0..127
    const int blk = blockIdx.x;   // 0..BN_BLOCKS-1
    const int chunk = (GN + BN_BLOCKS - 1) / BN_BLOCKS;
    const int r0 = blk * chunk;
    const int r1 = (r0 + chunk < GN) ? (r0 + chunk) : GN;
    float s = 0.0f, q = 0.0f;
    for (int r = r0; r < r1; ++r) {
        float v = h[(size_t)r * GD + col];
        s += v;
        q += v * v;
    }
    partial[(size_t)blk * 2 * GD + col] = s;
    partial[(size_t)blk * 2 * GD + GD + col] = q;
}

// ---------------------------------------------------------------------------
// BatchNorm pass 2: fold partials (fixed order) -> scale/shift per column.
// scaleshift[0:128] = g * rsqrt(var+eps); scaleshift[128:256] = b - mu*scale
// ---------------------------------------------------------------------------
__global__ __launch_bounds__(128) void bn_finalize_kernel(
    const float* __restrict__ partial, const float* __restrict__ g,
    const float* __restrict__ b, float* __restrict__ scaleshift) {
    const int col = threadIdx.x;
    float s = 0.0f, q = 0.0f;
    for (int blk = 0; blk < BN_BLOCKS; ++blk) {
        s += partial[(size_t)blk * 2 * GD + col];
        q += partial[(size_t)blk * 2 * GD + GD + col];
    }
    const float inv_n = 1.0f / (float)GN;
    const float mu = s * inv_n;
    const float var = q * inv_n - mu * mu;
    const float sc = g[col] * rsqrtf(var + 1e-5f);
    scaleshift[col] = sc;
    scaleshift[GD + col] = b[col] - mu * sc;
}

// ---------------------------------------------------------------------------
// x_out = x_in + relu(h * scale + shift)
// ---------------------------------------------------------------------------
__global__ void bn_relu_residual_kernel(const float* __restrict__ h,
                                        const float* __restrict__ scaleshift,
                                        const float* __restrict__ xin,
                                        float* __restrict__ xout) {
    size_t total = (size_t)GN * GD;
    for (size_t i = (size_t)blockIdx.x * blockDim.x + threadIdx.x; i < total;
         i += (size_t)gridDim.x * blockDim.x) {
        const int col = (int)(i & (GD - 1));
        float v = h[i] * scaleshift[col] + scaleshift[GD + col];
        v = fmaxf(v, 0.0f);
        xout[i] = xin[i] + v;
    }
}

// ---------------------------------------------------------------------------
// Final linear: out[N,47] = x @ lin_W^T + lin_b, WMMA tiles 16x16, 3 col tiles.
// Out-of-range column (47) handled by zeroing the B fragment (EXEC stays full).
// ---------------------------------------------------------------------------
__global__ __launch_bounds__(256) void final_linear_kernel(
    const float* __restrict__ x, const float* __restrict__ W,
    const float* __restrict__ bias, float* __restrict__ out) {
    const int NT_ROW = GN / 16;  // 6250
    const int lane = threadIdx.x & 31;
    const int wave = threadIdx.x >> 5;
    const int t = blockIdx.x * 8 + wave;
    if (t >= NT_ROW * 3) return;  // wave-uniform guard
    const int tr = t / 3;
    const int tc = t - tr * 3;
    const int row0 = tr * 16;
    const int col0 = tc * 16;
    const bool hi = lane >= 16;
    const int l16 = lane & 15;
    const int m = row0 + l16;
    const int koff = hi ? 2 : 0;
    const int n = col0 + l16;
    const float mask = (n < GC) ? 1.0f : 0.0f;
    const int nc = (n < GC) ? n : (GC - 1);

    v8f acc = {};
    const float* arow = x + (size_t)m * GD;
    const float* brow = W + (size_t)nc * GD;
    for (int k0 = 0; k0 < GD; k0 += 4) {
        v2f a = *(const v2f*)(arow + k0 + koff);
        v2f b = *(const v2f*)(brow + k0 + koff);
        b *= mask;
        acc = __builtin_amdgcn_wmma_f32_16x16x4_f32(false, a, false, b,
                                                    (short)0, acc, false,
                                                    false);
    }
#pragma unroll
    for (int v = 0; v < 8; ++v) {
        const int r = row0 + v + (hi ? 8 : 0);
        const int col = col0 + l16;
        if (col < GC) out[(size_t)r * GC + col] = acc[v] + bias[col];
    }
}

// ---------------------------------------------------------------------------
// Host-side launch
// ---------------------------------------------------------------------------
extern "C" void kernel_launch(void* const* d_in, const int* in_sizes, int n_in,
                              void* d_out, int out_size, void* d_ws,
                              size_t ws_size, hipStream_t stream) {
    const float* x_in = (const float*)d_in[0];
    const int* ei = (const int*)d_in[1];
    const int* src = ei;          // edge_index[0]
    const int* dst = ei + GE;     // edge_index[1]

    const float* Wl[3] = {(const float*)d_in[2], (const float*)d_in[7],
                          (const float*)d_in[12]};
    const float* bl[3] = {(const float*)d_in[3], (const float*)d_in[8],
                          (const float*)d_in[13]};
    const float* Wr[3] = {(const float*)d_in[4], (const float*)d_in[9],
                          (const float*)d_in[14]};
    const float* bg[3] = {(const float*)d_in[5], (const float*)d_in[10],
                          (const float*)d_in[15]};
    const float* bb[3] = {(const float*)d_in[6], (const float*)d_in[11],
                          (const float*)d_in[16]};
    const float* linW = (const float*)d_in[17];
    const float* linb = (const float*)d_in[18];
    float* out = (float*)d_out;

    // Workspace layout (floats)
    float* agg = (float*)d_ws;                       // N*D
    float* h = agg + (size_t)GN * GD;                // N*D
    float* xbuf = h + (size_t)GN * GD;               // N*D
    float* deg = xbuf + (size_t)GN * GD;             // N (becomes rdeg)
    float* partial = deg + GN;                       // BN_BLOCKS*2*D
    float* scsh = partial + (size_t)BN_BLOCKS * 2 * GD;  // 2*D

    const int NT = GN / 16;                   // 6250 row tiles
    const int gemm_grid = (NT + 7) / 8;       // 8 waves / block
    const int fin_tiles = NT * 3;
    const int fin_grid = (fin_tiles + 7) / 8;

    const float* xcur = x_in;
    for (int L = 0; L < 3; ++L) {
        zero_kernel<<<2048, 256, 0, stream>>>(agg, deg);
        scatter_kernel<<<GE / 8, 256, 0, stream>>>(xcur, src, dst, agg, deg);
        rdeg_kernel<<<(GN + 255) / 256, 256, 0, stream>>>(deg);
        sage_gemm_kernel<<<gemm_grid, 256, 0, stream>>>(xcur, agg, deg, Wl[L],
                                                        bl[L], Wr[L], h);
        bn_partial_kernel<<<BN_BLOCKS, 128, 0, stream>>>(h, partial);
        bn_finalize_kernel<<<1, 128, 0, stream>>>(partial, bg[L], bb[L], scsh);
        bn_relu_residual_kernel<<<2048, 256, 0, stream>>>(h, scsh, xcur, xbuf);
        xcur = xbuf;
    }
    final_linear_kernel<<<fin_grid, 256, 0, stream>>>(xcur, linW, linb, out);
}